// Conv1DISTFT_5488968204689
// MI455X (gfx1250) — compile-verified
//
#include <hip/hip_runtime.h>

// ISTFT on MI455X (gfx1250, wave32):
//   Stage 1: batched GEMM contrib[b,t,n] = sum_c X[b,c,t] * K[c,n]
//            via V_WMMA_F32_16X16X4_F32, 64x64 output tile per wave
//            (4x4 grid of 16x16 WMMA tiles, fp32 accumulate).
//   Stage 2: overlap-add + window^2 envelope normalize + crop, one output
//            sample per thread (deterministic, no float atomics).

#define N_FREQ   1025
#define N_FFT    2048
#define HOP      512
#define FRAMES   2048
#define ORIG_LEN ((FRAMES - 1) * HOP)   // 1048064
#define PAD      (N_FFT / 2)            // 1024

typedef float v2f __attribute__((ext_vector_type(2)));
typedef float v8f __attribute__((ext_vector_type(8)));

__device__ __forceinline__ v8f wmma_f32_16x16x4(v2f a, v2f b, v8f c) {
  // 8 args: (neg_a, A, neg_b, B, c_mod, C, reuse_a, reuse_b)
  return __builtin_amdgcn_wmma_f32_16x16x4_f32(false, a, false, b,
                                               (short)0, c, false, false);
}

// ---------------------------------------------------------------------------
// Stage 1: GEMM.  K-dim ordering is interleaved: c' = 2*f + {0=re,1=im}, so
// the A fragment (16x4 fp32: lanes 0-15 hold K={0,1}, lanes 16-31 K={2,3})
// is one contiguous 8-byte (re,im) load per lane from stft_repr[b,f,t,0:2].
// The B fragment reads kernel rows f (real) and f+1025 (imag).
// K total = 2050, padded to 2052 with one masked epilogue step.
// ---------------------------------------------------------------------------
__global__ __launch_bounds__(256)
void istft_gemm(const float* __restrict__ stft,   // [B,1025,2048,2]
                const float* __restrict__ kern,   // [2050,2048]
                float* __restrict__ contrib,      // [nb,2048,2048] in d_ws
                int b0) {
  const int lane = threadIdx.x & 31;
  const int wave = threadIdx.x >> 5;
  const int row  = lane & 15;      // M (A) / N (B) index within 16
  const int half = lane >> 4;      // K half: 0 -> K{0,1}, 1 -> K{2,3}

  // 1024 waves per batch: 32x32 tiles of 64x64
  const int waveIdx = blockIdx.x * 8 + wave;
  const int bLocal  = waveIdx >> 10;
  const int rr      = waveIdx & 1023;
  const int t0      = (rr >> 5) * 64;
  const int n0      = (rr & 31) * 64;
  const int b       = b0 + bLocal;

  const float* sb = stft + (size_t)b * N_FREQ * FRAMES * 2;

  v8f acc[4][4] = {};

  int tRow[4], nCol[4];
#pragma unroll
  for (int i = 0; i < 4; ++i) {
    tRow[i] = t0 + i * 16 + row;
    nCol[i] = n0 + i * 16 + row;
  }

  // Main K loop: c0 in [0,2048), f = c0/2 + half in [0,1024) -- always valid.
  for (int c0 = 0; c0 < 2048; c0 += 4) {
    const int f = (c0 >> 1) + half;
    v2f afrag[4], bfrag[4];
#pragma unroll
    for (int mi = 0; mi < 4; ++mi) {
      const size_t off = ((size_t)f * FRAMES + tRow[mi]) * 2;  // (re,im) pair
      afrag[mi] = *(const v2f*)(sb + off);
    }
#pragma unroll
    for (int ni = 0; ni < 4; ++ni) {
      v2f bv;
      bv.x = kern[(size_t)f * N_FFT + nCol[ni]];               // real row f
      bv.y = kern[(size_t)(f + N_FREQ) * N_FFT + nCol[ni]];    // imag row f
      bfrag[ni] = bv;
    }
#pragma unroll
    for (int mi = 0; mi < 4; ++mi)
#pragma unroll
      for (int ni = 0; ni < 4; ++ni)
        acc[mi][ni] = wmma_f32_16x16x4(afrag[mi], bfrag[ni], acc[mi][ni]);
  }

  // Epilogue step c0 = 2048: f = 1024 (valid, last freq) for half==0,
  // f = 1025 (pad) for half==1 -> mask to zero. EXEC stays all-ones.
  {
    const float m = (half == 0) ? 1.0f : 0.0f;
    const int fc = 1024;  // clamped safe index
    v2f afrag[4], bfrag[4];
#pragma unroll
    for (int mi = 0; mi < 4; ++mi) {
      const size_t off = ((size_t)fc * FRAMES + tRow[mi]) * 2;
      v2f av = *(const v2f*)(sb + off);
      afrag[mi] = av * m;
    }
#pragma unroll
    for (int ni = 0; ni < 4; ++ni) {
      v2f bv;
      bv.x = kern[(size_t)fc * N_FFT + nCol[ni]] * m;
      bv.y = kern[(size_t)(fc + N_FREQ) * N_FFT + nCol[ni]] * m;
      bfrag[ni] = bv;
    }
#pragma unroll
    for (int mi = 0; mi < 4; ++mi)
#pragma unroll
      for (int ni = 0; ni < 4; ++ni)
        acc[mi][ni] = wmma_f32_16x16x4(afrag[mi], bfrag[ni], acc[mi][ni]);
  }

  // Store: C/D layout is VGPR e -> M = e + 8*half, lane row -> N.
  float* cb = contrib + (size_t)bLocal * FRAMES * N_FFT;
#pragma unroll
  for (int mi = 0; mi < 4; ++mi) {
#pragma unroll
    for (int ni = 0; ni < 4; ++ni) {
      const int n = nCol[ni];
#pragma unroll
      for (int e = 0; e < 8; ++e) {
        const int t = t0 + mi * 16 + e + half * 8;
        cb[(size_t)t * N_FFT + n] = acc[mi][ni][e];
      }
    }
  }
}

// ---------------------------------------------------------------------------
// Stage 2: overlap-add, envelope normalize, crop. One output sample/thread:
// deterministic fixed-order sum over the <=4 overlapping frames.
// ---------------------------------------------------------------------------
__global__ __launch_bounds__(256)
void istft_oa(const float* __restrict__ contrib,  // [nb,2048,2048]
              const float* __restrict__ wsq,      // [2048]
              float* __restrict__ out,            // [B,ORIG_LEN]
              int b0, int nb) {
  const long long idx   = (long long)blockIdx.x * 256 + threadIdx.x;
  const long long total = (long long)nb * ORIG_LEN;
  if (idx >= total) return;

  const int bLocal = (int)(idx / ORIG_LEN);
  const int i      = (int)(idx % ORIG_LEN);
  const int s      = i + PAD;  // position in uncropped signal

  // frames t with 0 <= s - 512*t <= 2047
  const int tmin = (s >= N_FFT - 1) ? ((s - (N_FFT - 1) + (HOP - 1)) >> 9) : 0;
  int tmax = s >> 9;
  if (tmax > FRAMES - 1) tmax = FRAMES - 1;

  const float* cb = contrib + (size_t)bLocal * FRAMES * N_FFT;
  float num = 0.0f, den = 0.0f;
  for (int t = tmin; t <= tmax; ++t) {
    const int n = s - (t << 9);
    num += cb[(size_t)t * N_FFT + n];
    den += wsq[n];
  }
  out[(size_t)(b0 + bLocal) * ORIG_LEN + i] = num / fmaxf(den, 1e-8f);
}

// ---------------------------------------------------------------------------
extern "C" void kernel_launch(void* const* d_in, const int* in_sizes, int n_in,
                              void* d_out, int out_size, void* d_ws, size_t ws_size,
                              hipStream_t stream) {
  const float* stft = (const float*)d_in[0];  // [B,1025,2048,2] f32
  const float* kern = (const float*)d_in[1];  // [2050,2048]     f32
  const float* wsq  = (const float*)d_in[2];  // [2048]          f32
  float* out = (float*)d_out;                 // [B,1048064]     f32

  const int B = in_sizes[0] / (N_FREQ * FRAMES * 2);  // 8
  const size_t perBatch = (size_t)FRAMES * N_FFT * sizeof(float);  // 16.8 MB
  int chunk = (int)(ws_size / perBatch);
  if (chunk < 1) chunk = 1;   // assume ws >= one batch of contrib
  if (chunk > B) chunk = B;

  for (int b0 = 0; b0 < B; b0 += chunk) {
    const int nb = (B - b0 < chunk) ? (B - b0) : chunk;
    // nb * 1024 waves, 8 waves (256 threads) per block
    const int gemmBlocks = nb * 128;
    hipLaunchKernelGGL(istft_gemm, dim3(gemmBlocks), dim3(256), 0, stream,
                       stft, kern, (float*)d_ws, b0);
    const long long totalOut = (long long)nb * ORIG_LEN;
    const int oaBlocks = (int)((totalOut + 255) / 256);
    hipLaunchKernelGGL(istft_oa, dim3(oaBlocks), dim3(256), 0, stream,
                       (const float*)d_ws, wsq, out, b0, nb);
  }
}